// BoxFilter_52072183497292
// MI455X (gfx1250) — compile-verified
//
#include <hip/hip_runtime.h>

typedef __attribute__((ext_vector_type(2))) float v2f;
typedef __attribute__((ext_vector_type(8))) float v8f;

#define IMG_W 512
#define IMG_H 512
#define RAD   4          // r = 4  ->  9x9 box
// One wave32 computes one 16x16 output tile.
// Input region needed: 24x24 (tile + 8px halo), zero-padded at image edges.

__global__ __launch_bounds__(32)
void box9_wmma_kernel(const float* __restrict__ x, float* __restrict__ out)
{
    // LDS staging: Xs = zero-padded 24x24 input region (rows 24..31 forced to 0
    // so the second WMMA row-group reads zeros). Ts = intermediate T (24x16,
    // stored in 32 rows), used to convert D-fragment layout -> B-fragment layout.
    __shared__ float Xs[32][33];   // stride 33: conflict-free column walks
    __shared__ float Ts[32][17];

    const int lane = threadIdx.x;     // 0..31
    const int n    = lane & 15;       // N column / M row index within fragment
    const int kh   = lane >> 4;       // 0: lanes 0-15 (K lo half), 1: lanes 16-31 (K hi half)

    const int bid = blockIdx.x;
    const int tx  = bid & 31;                 // 512/16 = 32 tiles per row
    const int ty  = (bid >> 5) & 31;
    const int pl  = bid >> 10;                // plane index (N*C = 128)

    const float* __restrict__ src = x   + (size_t)pl * (IMG_H * IMG_W);
    float*       __restrict__ dst = out + (size_t)pl * (IMG_H * IMG_W);

    const int x0 = tx * 16;
    const int y0 = ty * 16;

    // ---- Stage 24 rows x 32 cols of input into LDS (coalesced, zero-padded) ----
    const int gx = x0 - RAD + lane;
    const bool cx = (gx >= 0) && (gx < IMG_W);
    #pragma unroll 4
    for (int j = 0; j < 24; ++j) {
        const int gy = y0 - RAD + j;
        float v = 0.0f;
        if (cx && gy >= 0 && gy < IMG_H) v = src[gy * IMG_W + gx];
        Xs[j][lane] = v;
    }
    #pragma unroll
    for (int j = 24; j < 32; ++j) Xs[j][lane] = 0.0f;
    __syncthreads();   // single-wave WG: LDS ordering only

    // ---- Pass 1: T(24x16) = In(24x24) x Sh(24x16),  Sh[i][X] = (X <= i <= X+8)
    // Two M-groups of 16 rows; K = 24 -> 6 slices of K=4.
    v8f t0 = {}, t1 = {};
    #pragma unroll
    for (int s = 0; s < 6; ++s) {
        const int j0 = 4 * s + 2 * kh;   // K index for A/B vgpr0
        const int j1 = j0 + 1;           // K index for A/B vgpr1
        v2f b;
        b.x = (n <= j0 && j0 <= n + 8) ? 1.0f : 0.0f;
        b.y = (n <= j1 && j1 <= n + 8) ? 1.0f : 0.0f;

        v2f a0, a1;
        a0.x = Xs[n][j0];        a0.y = Xs[n][j1];        // M rows 0..15
        a1.x = Xs[16 + n][j0];   a1.y = Xs[16 + n][j1];   // M rows 16..31 (24..31 = 0)

        t0 = __builtin_amdgcn_wmma_f32_16x16x4_f32(false, a0, false, b, (short)0, t0, false, false);
        t1 = __builtin_amdgcn_wmma_f32_16x16x4_f32(false, a1, false, b, (short)0, t1, false, false);
    }

    // ---- D-fragment -> LDS (rows M = r + 8*kh per group) ----
    #pragma unroll
    for (int r = 0; r < 8; ++r) {
        Ts[r + 8 * kh][n]      = t0[r];
        Ts[16 + r + 8 * kh][n] = t1[r];
    }
    __syncthreads();

    // ---- Pass 2: Out(16x16) = Sv(16x24) x T(24x16),  Sv[Y][j] = (Y <= j <= Y+8)
    v8f o = {};
    #pragma unroll
    for (int s = 0; s < 6; ++s) {
        const int j0 = 4 * s + 2 * kh;
        const int j1 = j0 + 1;
        v2f a;                                   // Sv fragment: M row = n
        a.x = (n <= j0 && j0 <= n + 8) ? 1.0f : 0.0f;
        a.y = (n <= j1 && j1 <= n + 8) ? 1.0f : 0.0f;
        v2f b;                                   // T as B operand: K row = j, N col = n
        b.x = Ts[j0][n];
        b.y = Ts[j1][n];
        o = __builtin_amdgcn_wmma_f32_16x16x4_f32(false, a, false, b, (short)0, o, false, false);
    }

    // ---- Scale by 1/81 and store (two 64B segments per store instruction) ----
    const float sc = 1.0f / 81.0f;
    #pragma unroll
    for (int r = 0; r < 8; ++r) {
        const int oy = y0 + r + 8 * kh;
        dst[oy * IMG_W + x0 + n] = o[r] * sc;
    }
}

extern "C" void kernel_launch(void* const* d_in, const int* in_sizes, int n_in,
                              void* d_out, int out_size, void* d_ws, size_t ws_size,
                              hipStream_t stream)
{
    (void)in_sizes; (void)n_in; (void)d_ws; (void)ws_size; (void)out_size;
    const float* x = (const float*)d_in[0];   // (4, 32, 512, 512) fp32
    float* out = (float*)d_out;

    // 128 planes * 32*32 tiles = 131072 workgroups of one wave32 each.
    const int planes = 4 * 32;
    dim3 grid(planes * (IMG_H / 16) * (IMG_W / 16));
    dim3 block(32);
    box9_wmma_kernel<<<grid, block, 0, stream>>>(x, out);
}